// GraphSAGE_39840116638555
// MI455X (gfx1250) — compile-verified
//
#include <hip/hip_runtime.h>

// ---------------- problem constants (match reference) ----------------
#define N_NODES 50000
#define EDGES   1600000
#define DIM     256
#define HID     256
#define SCAN_B  ((N_NODES + 255) / 256)   // 196 scan blocks

typedef __attribute__((ext_vector_type(2))) float v2f;
typedef __attribute__((ext_vector_type(8))) float v8f;

// ---------------- zero fill ----------------
__global__ void zero_f32(float* __restrict__ p, long n) {
    long i = (long)blockIdx.x * blockDim.x + threadIdx.x;
    if (i < n) p[i] = 0.0f;
}
__global__ void zero_i32(int* __restrict__ p, long n) {
    long i = (long)blockIdx.x * blockDim.x + threadIdx.x;
    if (i < n) p[i] = 0;
}

// ---------------- CSR build: count -> scan -> fill ----------------
__global__ void count_edges(const int* __restrict__ dst, int* __restrict__ cnt) {
    long e = (long)blockIdx.x * blockDim.x + threadIdx.x;
    if (e < EDGES) atomicAdd(&cnt[dst[e]], 1);
}

// Phase A: per-block inclusive scan of cnt chunk -> row_ptr[i+1]; block totals -> bsum
__global__ void scan_phaseA(const int* __restrict__ cnt, int* __restrict__ row_ptr,
                            int* __restrict__ bsum) {
    __shared__ int sm[256];
    int i = blockIdx.x * 256 + threadIdx.x;
    int v = (i < N_NODES) ? cnt[i] : 0;
    sm[threadIdx.x] = v;
    __syncthreads();
    for (int off = 1; off < 256; off <<= 1) {
        int add = (threadIdx.x >= off) ? sm[threadIdx.x - off] : 0;
        __syncthreads();
        sm[threadIdx.x] += add;
        __syncthreads();
    }
    if (i < N_NODES) row_ptr[i + 1] = sm[threadIdx.x];
    if (threadIdx.x == 255) bsum[blockIdx.x] = sm[255];
}

// Phase B: single-block inclusive scan of the block totals
__global__ void scan_phaseB(int* __restrict__ bsum) {
    __shared__ int sm[256];
    int v = (threadIdx.x < SCAN_B) ? bsum[threadIdx.x] : 0;
    sm[threadIdx.x] = v;
    __syncthreads();
    for (int off = 1; off < 256; off <<= 1) {
        int add = (threadIdx.x >= off) ? sm[threadIdx.x - off] : 0;
        __syncthreads();
        sm[threadIdx.x] += add;
        __syncthreads();
    }
    if (threadIdx.x < SCAN_B) bsum[threadIdx.x] = sm[threadIdx.x];
}

// Phase C: add block offsets in place; set row_ptr[0] = 0
__global__ void scan_phaseC(int* __restrict__ row_ptr, const int* __restrict__ bsum) {
    int i = blockIdx.x * 256 + threadIdx.x;
    if (i == 0) row_ptr[0] = 0;
    if (i < N_NODES && blockIdx.x > 0) row_ptr[i + 1] += bsum[blockIdx.x - 1];
}

__global__ void fill_adj(const int* __restrict__ src, const int* __restrict__ dst,
                         const int* __restrict__ row_ptr, int* __restrict__ cursor,
                         int* __restrict__ nbr) {
    long e = (long)blockIdx.x * blockDim.x + threadIdx.x;
    if (e >= EDGES) return;
    int d = dst[e];
    int pos = atomicAdd(&cursor[d], 1);
    nbr[row_ptr[d] + pos] = src[e];
}

// ---------------- 256x256 transpose (weights -> column-contiguous) ----------------
__global__ __launch_bounds__(256)
void transpose256(const float* __restrict__ in, float* __restrict__ outT) {
    __shared__ float tile[16][17];
    const int tx = threadIdx.x & 15;
    const int ty = threadIdx.x >> 4;
    tile[ty][tx] = in[(size_t)(blockIdx.y * 16 + ty) * HID + blockIdx.x * 16 + tx];
    __syncthreads();
    outT[(size_t)(blockIdx.x * 16 + ty) * DIM + blockIdx.y * 16 + tx] = tile[tx][ty];
}

// ---------------- neighbor mean via CSR gather (no atomics, b128 loads) ----------------
// 4 nodes per block; 64 threads per node, each owning 4 consecutive features.
__global__ __launch_bounds__(256)
void gather_mean(const float* __restrict__ h, const int* __restrict__ remap,
                 const int* __restrict__ row_ptr, const int* __restrict__ nbr,
                 float* __restrict__ hneigh) {
    const int g = threadIdx.x >> 6;          // node subgroup 0..3
    const int t = threadIdx.x & 63;          // feature group (4 floats)
    const int n = blockIdx.x * 4 + g;        // N divisible by 4 (50000/4 = 12500)
    const int beg = row_ptr[n];
    const int end = row_ptr[n + 1];
    float4 acc = make_float4(0.f, 0.f, 0.f, 0.f);
    #pragma unroll 2
    for (int i = beg; i < end; ++i) {
        int s = nbr[i];
        if (remap) s = remap[s];
        const float4 v = *(const float4*)(h + (size_t)s * DIM + t * 4);
        acc.x += v.x; acc.y += v.y; acc.z += v.z; acc.w += v.w;
    }
    const float invd = 1.0f / fmaxf((float)(end - beg), 1.0f);
    acc.x *= invd; acc.y *= invd; acc.z *= invd; acc.w *= invd;
    *(float4*)(hneigh + (size_t)n * DIM + t * 4) = acc;
}

// ---------------- fused SAGE layer GEMM via V_WMMA_F32_16X16X4_F32 ----------------
// out[n, j] = act( h[n,:]·WselfT[j,:] + hneigh[n,:]·WneighT[j,:] + b[j] )
// Weights pre-transposed: WT[j, k], so B fetches are contiguous b64 per lane.
// One wave32 computes one 16x16 output tile. A (16x4 f32): lanes 0-15 hold rows
// M=0..15 with K={0,1}; lanes 16-31 hold the same rows with K={2,3}. B mirrors
// with N in place of M.
template <bool RELU>
__global__ __launch_bounds__(256)
void sage_gemm(const float* __restrict__ h, const int* __restrict__ remap,
               const float* __restrict__ hneigh,
               const float* __restrict__ WselfT, const float* __restrict__ WneighT,
               const float* __restrict__ bias, float* __restrict__ out) {
    const int lane = threadIdx.x & 31;
    const int wave = blockIdx.x * (blockDim.x >> 5) + (threadIdx.x >> 5);
    const int jt = wave & (HID / 16 - 1);   // 16 column tiles
    const int mt = wave >> 4;               // 3125 row tiles (exact)
    const int j0 = jt << 4;
    const int m0 = mt << 4;
    const int l    = lane & 15;
    const int half = lane >> 4;             // 0: K={0,1}, 1: K={2,3}
    const int ko   = half << 1;

    const int rowN = m0 + l;
    const int rowH = remap ? remap[rowN] : rowN;
    const float* __restrict__ aSelf  = h      + (size_t)rowH * DIM;
    const float* __restrict__ aNeigh = hneigh + (size_t)rowN * DIM;
    const float* __restrict__ wsp = WselfT  + (size_t)(j0 + l) * DIM;  // row j0+l of WT
    const float* __restrict__ wnp = WneighT + (size_t)(j0 + l) * DIM;

    v8f acc = {};
    #pragma unroll 4
    for (int kk = 0; kk < DIM; kk += 4) {
        const int ka = kk + ko;
        // --- self-feature contribution ---
        v2f a  = *(const v2f*)(aSelf + ka);
        v2f b  = *(const v2f*)(wsp + ka);
        acc = __builtin_amdgcn_wmma_f32_16x16x4_f32(false, a, false, b,
                                                    (short)0, acc, false, false);
        // --- neighbor-mean contribution ---
        v2f a2 = *(const v2f*)(aNeigh + ka);
        v2f b2 = *(const v2f*)(wnp + ka);
        acc = __builtin_amdgcn_wmma_f32_16x16x4_f32(false, a2, false, b2,
                                                    (short)0, acc, false, false);
    }

    // Epilogue: lane -> column j0+l; VGPR v -> row m0 + v + half*8.
    const float bj = bias[j0 + l];
    float* __restrict__ o = out + (size_t)(m0 + half * 8) * HID + j0 + l;
    #pragma unroll
    for (int v = 0; v < 8; ++v) {
        float r = acc[v] + bj;
        if (RELU) r = fmaxf(r, 0.0f);
        o[(size_t)v * HID] = r;
    }
}

// ---------------- global Frobenius norm ----------------
__global__ void reduce_sq(const float* __restrict__ x, long n, float* __restrict__ accum) {
    __shared__ float sm[256];
    float s = 0.0f;
    for (long i = (long)blockIdx.x * blockDim.x + threadIdx.x; i < n;
         i += (long)gridDim.x * blockDim.x) {
        float v = x[i];
        s += v * v;
    }
    sm[threadIdx.x] = s;
    __syncthreads();
    for (int off = 128; off > 0; off >>= 1) {
        if (threadIdx.x < off) sm[threadIdx.x] += sm[threadIdx.x + off];
        __syncthreads();
    }
    if (threadIdx.x == 0) atomicAdd(accum, sm[0]);
}

__global__ void scale_out(float* __restrict__ x, long n, const float* __restrict__ accum) {
    long i = (long)blockIdx.x * blockDim.x + threadIdx.x;
    if (i < n) x[i] *= rsqrtf(*accum);
}

// ---------------- launch ----------------
extern "C" void kernel_launch(void* const* d_in, const int* in_sizes, int n_in,
                              void* d_out, int out_size, void* d_ws, size_t ws_size,
                              hipStream_t stream) {
    const int*   input_nodes = (const int*)d_in[0];
    const int*   esrc = (const int*)d_in[1];
    const int*   edst = (const int*)d_in[2];
    const float* emb  = (const float*)d_in[3];
    const float* Ws0  = (const float*)d_in[4];
    const float* Wn0  = (const float*)d_in[5];
    const float* b0   = (const float*)d_in[6];
    const float* Ws1  = (const float*)d_in[7];
    const float* Wn1  = (const float*)d_in[8];
    const float* b1   = (const float*)d_in[9];

    // workspace layout (floats first to keep 16B alignment for b128/b64 access)
    float* hneigh  = (float*)d_ws;                        // N*DIM
    float* h1      = hneigh + (size_t)N_NODES * DIM;      // N*HID
    float* WsT0    = h1 + (size_t)N_NODES * HID;          // DIM*HID
    float* WnT0    = WsT0 + (size_t)DIM * HID;            // DIM*HID
    float* WsT1    = WnT0 + (size_t)DIM * HID;            // DIM*HID
    float* WnT1    = WsT1 + (size_t)DIM * HID;            // DIM*HID
    float* nrm     = WnT1 + (size_t)DIM * HID;            // 1
    int*   cnt     = (int*)(nrm + 1);                     // N (count, then cursor)
    int*   row_ptr = cnt + N_NODES;                       // N+1
    int*   bsum    = row_ptr + (N_NODES + 1);             // 256
    int*   nbr     = bsum + 256;                          // E
    float* out     = (float*)d_out;

    const unsigned egrid = (unsigned)((EDGES + 255) / 256);
    const unsigned ngrid = (unsigned)((N_NODES + 255) / 256);
    const int gemmBlocks = (N_NODES / 16) * (HID / 16) / 8;   // 8 waves / 256-thr block
    const dim3 tgrid(HID / 16, DIM / 16);

    // ---- weight transposes (column-contiguous B operands) ----
    transpose256<<<tgrid, 256, 0, stream>>>(Ws0, WsT0);
    transpose256<<<tgrid, 256, 0, stream>>>(Wn0, WnT0);
    transpose256<<<tgrid, 256, 0, stream>>>(Ws1, WsT1);
    transpose256<<<tgrid, 256, 0, stream>>>(Wn1, WnT1);

    // ---- CSR build (same graph for both layers) ----
    zero_i32<<<ngrid, 256, 0, stream>>>(cnt, N_NODES);
    count_edges<<<egrid, 256, 0, stream>>>(edst, cnt);
    scan_phaseA<<<SCAN_B, 256, 0, stream>>>(cnt, row_ptr, bsum);
    scan_phaseB<<<1, 256, 0, stream>>>(bsum);
    scan_phaseC<<<SCAN_B, 256, 0, stream>>>(row_ptr, bsum);
    zero_i32<<<ngrid, 256, 0, stream>>>(cnt, N_NODES);        // reuse as cursor
    fill_adj<<<egrid, 256, 0, stream>>>(esrc, edst, row_ptr, cnt, nbr);

    // ---- layer 0 ----
    gather_mean<<<N_NODES / 4, 256, 0, stream>>>(emb, input_nodes, row_ptr, nbr, hneigh);
    sage_gemm<true><<<gemmBlocks, 256, 0, stream>>>(emb, input_nodes, hneigh,
                                                    WsT0, WnT0, b0, h1);

    // ---- layer 1 ----
    gather_mean<<<N_NODES / 4, 256, 0, stream>>>(h1, nullptr, row_ptr, nbr, hneigh);
    sage_gemm<false><<<gemmBlocks, 256, 0, stream>>>(h1, nullptr, hneigh,
                                                     WsT1, WnT1, b1, out);

    // ---- global Frobenius norm scale ----
    zero_f32<<<1, 32, 0, stream>>>(nrm, 1);
    const long on = (long)N_NODES * HID;
    reduce_sq<<<1024, 256, 0, stream>>>(out, on, nrm);
    scale_out<<<(unsigned)((on + 255) / 256), 256, 0, stream>>>(out, on, nrm);
}